// GNNProcessor_86672440033638
// MI455X (gfx1250) — compile-verified
//
#include <hip/hip_runtime.h>
#include <hip/hip_bf16.h>

// ---------------------------------------------------------------------------
// GraphConv x2 for MI455X (gfx1250, wave32).
//   Layer: out = agg @ w_rel + x @ w_root + b ; agg = segment_sum(x[src], dst)
// Edge phase: L2-resident gather + native f32 atomics.
// Dense phase: V_WMMA_F32_16X16X4_F32, one wave per 16-row tile, fused bias
// (+ReLU for layer 1), both GEMMs accumulated into one C fragment.
// ---------------------------------------------------------------------------

typedef __attribute__((ext_vector_type(2))) float v2f;
typedef __attribute__((ext_vector_type(8))) float v8f;

#define DF 48  // D_IN == D_H == D_OUT == 48

// ---------------- zero the aggregation buffer (float4 stores) --------------
__global__ void zero_f4(float4* __restrict__ p, int n4) {
    int gid = blockIdx.x * blockDim.x + threadIdx.x;
    if (gid < n4) p[gid] = make_float4(0.f, 0.f, 0.f, 0.f);
}

// ---------------- edge scatter: agg[dst] += feat[src] ----------------------
// One thread per (edge, feature). Gathers hit L2 (feature table is 19.2MB);
// adds are native global_atomic_add_f32 (no return -> STOREcnt path).
__global__ void scatter_add(const float* __restrict__ feat,
                            const int* __restrict__ src,
                            const int* __restrict__ dst,
                            float* __restrict__ agg,
                            long long total) {
    long long gid = (long long)blockIdx.x * blockDim.x + threadIdx.x;
    if (gid >= total) return;
    int e = (int)(gid / DF);
    int f = (int)(gid - (long long)e * DF);
    int s = src[e];
    int d = dst[e];
    unsafeAtomicAdd(&agg[(long long)d * DF + f], feat[(long long)s * DF + f]);
}

// ---------------- fused dual-GEMM + bias (+ReLU) via WMMA ------------------
// One wave32 per 16-row node tile. EXEC stays all-ones (no divergence before
// WMMA); row clamping for a ragged tail is done via index clamp + predicated
// stores only (N=100000 is divisible by 16, so the clamp is a no-op here).
//
// ISA layouts (CDNA5 ISA §7.12.2):
//   A 16x4 f32 : lane<16 holds row (lane), K={4k,4k+1}; lane>=16 K={4k+2,4k+3}
//   B 4x16 f32 : lane&15 = column, (lane>>4) selects K pair {0,1}/{2,3}
//   C/D 16x16  : VGPR v -> M = v + 8*(lane>>4), N = lane&15
template <bool RELU>
__global__ __launch_bounds__(32)
void gemm_dual_wmma(const float* __restrict__ agg,   // [N,48] aggregated
                    const float* __restrict__ xin,   // [N,48] self features
                    const float* __restrict__ w_rel, // [48,48] row-major
                    const float* __restrict__ w_root,// [48,48] row-major
                    const float* __restrict__ bias,  // [48]
                    float* __restrict__ out,         // [N,48]
                    int n_nodes) {
    const int lane = threadIdx.x;   // 0..31
    const int hi   = lane >> 4;     // 0/1 : selects K sub-pair / M offset
    const int lo   = lane & 15;
    const int row0 = blockIdx.x * 16;

    // ---- load A fragments for both operands (12 K-steps of 4) ----
    const int arow = min(row0 + lo, n_nodes - 1);   // clamp, no divergence
    v2f a_agg[12], a_x[12];
#pragma unroll
    for (int k = 0; k < 12; ++k) {
        const int kk = 4 * k + 2 * hi;
        const long long base = (long long)arow * DF + kk;
        a_agg[k].x = agg[base];
        a_agg[k].y = agg[base + 1];
        a_x[k].x   = xin[base];
        a_x[k].y   = xin[base + 1];
    }

    // ---- 3 output column tiles of 16 ----
#pragma unroll
    for (int nt = 0; nt < 3; ++nt) {
        const int ncol = nt * 16 + lo;
        const float bv = bias[ncol];      // column-only -> same for all 8 regs
        v8f c;
#pragma unroll
        for (int i = 0; i < 8; ++i) c[i] = bv;

#pragma unroll
        for (int k = 0; k < 12; ++k) {
            const int krow = 4 * k + 2 * hi;
            v2f brel, broot;
            brel.x  = w_rel [(krow    ) * DF + ncol];
            brel.y  = w_rel [(krow + 1) * DF + ncol];
            broot.x = w_root[(krow    ) * DF + ncol];
            broot.y = w_root[(krow + 1) * DF + ncol];
            // c += a_agg * w_rel
            c = __builtin_amdgcn_wmma_f32_16x16x4_f32(
                    false, a_agg[k], false, brel, (short)0, c, false, false);
            // c += a_x * w_root
            c = __builtin_amdgcn_wmma_f32_16x16x4_f32(
                    false, a_x[k], false, broot, (short)0, c, false, false);
        }

        // ---- store D: VGPR v -> row (row0 + v + 8*hi), col ncol ----
#pragma unroll
        for (int v = 0; v < 8; ++v) {
            float r = c[v];
            if (RELU) r = fmaxf(r, 0.f);
            const int m = row0 + v + 8 * hi;
            if (m < n_nodes) out[(long long)m * DF + ncol] = r;
        }
    }
}

extern "C" void kernel_launch(void* const* d_in, const int* in_sizes, int n_in,
                              void* d_out, int out_size, void* d_ws, size_t ws_size,
                              hipStream_t stream) {
    const float* x       = (const float*)d_in[0];
    const int*   ei      = (const int*)  d_in[1];   // [2, E] int32
    const float* w_rel1  = (const float*)d_in[2];
    const float* w_root1 = (const float*)d_in[3];
    const float* b1      = (const float*)d_in[4];
    const float* w_rel2  = (const float*)d_in[5];
    const float* w_root2 = (const float*)d_in[6];
    const float* b2      = (const float*)d_in[7];
    float*       out     = (float*)d_out;

    const int N = in_sizes[0] / DF;        // 100000
    const int E = in_sizes[1] / 2;         // 1600000
    const int* src = ei;
    const int* dst = ei + E;

    float* agg = (float*)d_ws;             // [N,48]
    float* h   = agg + (size_t)N * DF;     // [N,48]

    const int nfeat   = N * DF;
    const int n4      = nfeat / 4;
    const int zblocks = (n4 + 255) / 256;

    const long long etotal = (long long)E * DF;
    const int eblocks = (int)((etotal + 255) / 256);

    const int mtiles = (N + 15) / 16;      // 6250

    // ---------------- layer 1 ----------------
    zero_f4<<<zblocks, 256, 0, stream>>>((float4*)agg, n4);
    scatter_add<<<eblocks, 256, 0, stream>>>(x, src, dst, agg, etotal);
    gemm_dual_wmma<true><<<mtiles, 32, 0, stream>>>(agg, x, w_rel1, w_root1, b1, h, N);

    // ---------------- layer 2 ----------------
    zero_f4<<<zblocks, 256, 0, stream>>>((float4*)agg, n4);
    scatter_add<<<eblocks, 256, 0, stream>>>(h, src, dst, agg, etotal);
    gemm_dual_wmma<false><<<mtiles, 32, 0, stream>>>(agg, h, w_rel2, w_root2, b2, out, N);
}